// VQViT_22119081574825
// MI455X (gfx1250) — compile-verified
//
#include <hip/hip_runtime.h>

typedef __attribute__((ext_vector_type(2))) float v2f;
typedef __attribute__((ext_vector_type(8))) float v8f;
typedef __attribute__((ext_vector_type(4))) unsigned int v4u;
typedef __attribute__((ext_vector_type(8))) int v8i;
typedef __attribute__((ext_vector_type(4))) int v4i;

#define N_E   8192
#define E_DIM 256
#define M_TOK 8192
#define BETA  0.25f
#define ROWSTRIDE 260   // padded LDS row stride in floats (bank-conflict-free b64)
#define NTILE 32        // codes per workgroup iteration (16 cols x 2 per wave)

// d_out layout (floats)
#define OUT_ZQ   0
#define OUT_LC   2097152
#define OUT_LB   2097153
#define OUT_IDX  2097154
#define OUT_UTIL 2105346
#define OUT_MEAN 2105347

// LDS byte offsets (dynamic LDS starts at 0; kernel has no static LDS)
#define ZL_FLOATS   (128 * ROWSTRIDE)
#define EL_FLOATS   (NTILE * ROWSTRIDE)
#define EL0_BYTEOFF (ZL_FLOATS * 4)
#define EL1_BYTEOFF ((ZL_FLOATS + EL_FLOATS) * 4)

// ---------------------------------------------------------------- kernel 1
// ||e||^2 per code, one wave per code.
__global__ __launch_bounds__(256) void vq_enorm(const float* __restrict__ emb,
                                                float* __restrict__ enorm) {
    const int lane = threadIdx.x & 31;
    const int wave = threadIdx.x >> 5;
    const int code = blockIdx.x * 8 + wave;
    const float* e = emb + (size_t)code * E_DIM;
    float s = 0.f;
#pragma unroll
    for (int i = 0; i < 8; ++i) { float v = e[lane + 32 * i]; s += v * v; }
#pragma unroll
    for (int m = 16; m > 0; m >>= 1) s += __shfl_xor(s, m, 32);
    if (lane == 0) enorm[code] = s;
}

// ---------------------------------------------------------------- TDM issue
// DMA one 32x256-f32 tile of emb (rows n0..n0+31) into LDS at lds_byteoff,
// hardware-padding each 256-DWORD row by 4 DWORDs -> 260-float LDS stride.
__device__ __forceinline__ void tdm_load_tile(const float* emb, int n0,
                                              unsigned int lds_byteoff) {
    unsigned long long ga = (unsigned long long)(uintptr_t)(emb + (size_t)n0 * E_DIM);
    v4u g0;
    g0.x = 1u;                                    // count=1, user descriptor
    g0.y = lds_byteoff;                           // lds_addr [63:32]
    g0.z = (unsigned int)ga;                      // global_addr [95:64]
    g0.w = (unsigned int)(ga >> 32) | 0x80000000u; // global_addr[56:32] | type=2

    v8i g1;
    g1[0] = (int)((2u << 16)      // data_size = 4 bytes
                | (1u << 20)      // pad_enable
                | (7u << 22)      // pad_interval: 256 DWORDs
                | (3u << 25));    // pad_amount: 4 DWORDs
    g1[1] = (int)(256u << 16);    // tensor_dim0 = 256        [79:48] lo16
    g1[2] = (int)(8192u << 16);   // tensor_dim1 = 8192       [111:80] lo16
    g1[3] = (int)(256u << 16);    // tile_dim0 = 256          [127:112]
    g1[4] = (int)NTILE;           // tile_dim1 = 32           [143:128]
    g1[5] = 256;                  // tensor_dim0_stride = 256 [207:160] lo32
    g1[6] = 0;
    g1[7] = 0;

    v4i gz4 = {0, 0, 0, 0};
    v8i gz8 = {0, 0, 0, 0, 0, 0, 0, 0};
    // 6-arg form (clang-23 / therock-10.0 headers on this toolchain)
    __builtin_amdgcn_tensor_load_to_lds(g0, g1, gz4, gz4, gz8, 0);
}

// ---------------------------------------------------------------- kernel 2
// Fused (-2 z @ emb^T + ||e||^2) + running argmin via f32 WMMA.
// 64 blocks x 256 threads; each block owns 128 token rows; each wave owns 16
// rows x 32 codes per iteration (two 16x16 WMMA tiles sharing the A fragment).
__global__ __launch_bounds__(256) void vq_argmin(const float* __restrict__ z,
                                                 const float* __restrict__ emb,
                                                 const float* __restrict__ enorm,
                                                 int* __restrict__ idxbuf,
                                                 float* __restrict__ idxf) {
    extern __shared__ float smem[];
    float* zL  = smem;                              // 128 * ROWSTRIDE, holds -2*z
    float* eL0 = smem + ZL_FLOATS;                  // emb tile double buffer
    float* eL1 = smem + ZL_FLOATS + EL_FLOATS;

    const int t    = threadIdx.x;
    const int lane = t & 31;
    const int wave = t >> 5;
    const int half = lane >> 4;    // selects K pair within 16x4 fragment
    const int lcol = lane & 15;    // M row (A) / N col (B,C,D)
    const int m0   = blockIdx.x * 128;

    // kick off DMA of the first emb tile while we stage z
    if (wave == 0) tdm_load_tile(emb, 0, EL0_BYTEOFF);

    // stage -2*z for this block's 128 rows (coalesced: thread t -> column t)
    for (int i = 0; i < 128; ++i)
        zL[i * ROWSTRIDE + t] = -2.0f * z[(size_t)(m0 + i) * E_DIM + t];

    float bv[8];
    int   bi[8];
#pragma unroll
    for (int j = 0; j < 8; ++j) { bv[j] = 3.4e38f; bi[j] = 0; }

    const int aoff = (wave * 16 + lcol) * ROWSTRIDE + 2 * half;  // A: row=lcol
    const int boff = lcol * ROWSTRIDE + 2 * half;                // B: col=lcol

    for (int nt = 0; nt < N_E / NTILE; ++nt) {
        const int n0 = nt * NTILE;
        __syncthreads();   // prior compute done: buffer (nt+1)&1 free; zL staged
        if (wave == 0) {
            if (nt + 1 < N_E / NTILE) {
                tdm_load_tile(emb, n0 + NTILE,
                              ((nt + 1) & 1) ? EL1_BYTEOFF : EL0_BYTEOFF);
                __builtin_amdgcn_s_wait_tensorcnt(1);  // tile nt has landed
            } else {
                __builtin_amdgcn_s_wait_tensorcnt(0);
            }
        }
        __syncthreads();   // tile nt visible to all waves

        const float* eb = (nt & 1) ? eL1 : eL0;

        // seed accumulators with ||e||^2 of each lane's two columns
        const float en0 = enorm[n0 + lcol];
        const float en1 = enorm[n0 + 16 + lcol];
        v8f c0 = { en0, en0, en0, en0, en0, en0, en0, en0 };
        v8f c1 = { en1, en1, en1, en1, en1, en1, en1, en1 };

        // D = sum_k (-2 z) * e + ||e||^2 == distance score (z-norm dropped)
#pragma unroll
        for (int k0 = 0; k0 < E_DIM; k0 += 4) {
            v2f a  = *(const v2f*)(zL + aoff + k0);
            v2f b0 = *(const v2f*)(eb + boff + k0);
            v2f b1 = *(const v2f*)(eb + boff + 16 * ROWSTRIDE + k0);
            c0 = __builtin_amdgcn_wmma_f32_16x16x4_f32(
                     false, a, false, b0, (short)0, c0, false, false);
            c1 = __builtin_amdgcn_wmma_f32_16x16x4_f32(
                     false, a, false, b1, (short)0, c1, false, false);
        }

        const int i0 = n0 + lcol;
        const int i1 = n0 + 16 + lcol;
#pragma unroll
        for (int j = 0; j < 8; ++j) {   // check lower indices first: ties keep first
            if (c0[j] < bv[j]) { bv[j] = c0[j]; bi[j] = i0; }
            if (c1[j] < bv[j]) { bv[j] = c1[j]; bi[j] = i1; }
        }
    }

    // reduce argmin across the 16 column lanes (masks stay within half-wave)
#pragma unroll
    for (int j = 0; j < 8; ++j) {
#pragma unroll
        for (int m = 8; m >= 1; m >>= 1) {
            float ov = __shfl_xor(bv[j], m, 32);
            int   oi = __shfl_xor(bi[j], m, 32);
            if (ov < bv[j] || (ov == bv[j] && oi < bi[j])) { bv[j] = ov; bi[j] = oi; }
        }
        if (lcol == 0) {   // lane 0 -> rows 0..7 of wave tile, lane 16 -> rows 8..15
            const int grow = m0 + wave * 16 + half * 8 + j;
            idxbuf[grow] = bi[j];
            idxf[grow]   = (float)bi[j];
        }
    }
}

// ---------------------------------------------------------------- kernel 3
// Gather z_q, write straight-through output, per-row partial of (z_q - z)^2.
__global__ __launch_bounds__(256) void vq_gather_loss(const float* __restrict__ z,
                                                      const float* __restrict__ emb,
                                                      const int* __restrict__ idxbuf,
                                                      float* __restrict__ out_zq,
                                                      float* __restrict__ partial) {
    const int m = blockIdx.x;
    const int t = threadIdx.x;
    const int idx = idxbuf[m];
    const float zq = emb[(size_t)idx * E_DIM + t];
    const float zv = z[(size_t)m * E_DIM + t];
    out_zq[(size_t)m * E_DIM + t] = zq;
    float d = zq - zv;
    d *= d;
    __shared__ float red[256];
    red[t] = d;
    __syncthreads();
    for (int off = 128; off > 0; off >>= 1) {
        if (t < off) red[t] += red[t + off];
        __syncthreads();
    }
    if (t == 0) partial[m] = red[0];
}

// ---------------------------------------------------------------- kernel 4
// Deterministic fixed-order reduction of partials + code_age stats + scalars.
__global__ __launch_bounds__(256) void vq_finalize(const float* __restrict__ partial,
                                                   const float* __restrict__ code_age,
                                                   float* __restrict__ d_out) {
    __shared__ float sdat[256];
    __shared__ float scnt[256];
    __shared__ float ssum[256];
    const int t = threadIdx.x;
    float s = 0.f;
    for (int i = 0; i < M_TOK / 256; ++i) s += partial[t + 256 * i];
    float cnt = 0.f, asum = 0.f;
    for (int i = 0; i < N_E / 256; ++i) {
        float a = code_age[t + 256 * i];
        cnt += (a < 128.0f) ? 1.0f : 0.0f;   // DEAD_LIMIT//2
        asum += a;
    }
    sdat[t] = s; scnt[t] = cnt; ssum[t] = asum;
    __syncthreads();
    for (int off = 128; off > 0; off >>= 1) {
        if (t < off) {
            sdat[t] += sdat[t + off];
            scnt[t] += scnt[t + off];
            ssum[t] += ssum[t + off];
        }
        __syncthreads();
    }
    if (t == 0) {
        const float lc = sdat[0] / (float)((size_t)M_TOK * E_DIM);
        d_out[OUT_LC]   = lc;
        d_out[OUT_LB]   = BETA * lc;       // numerically identical mean
        d_out[OUT_UTIL] = scnt[0] / (float)N_E;
        d_out[OUT_MEAN] = ssum[0] / (float)N_E;
    }
}

// ----------------------------------------------------------------
extern "C" void kernel_launch(void* const* d_in, const int* in_sizes, int n_in,
                              void* d_out, int out_size, void* d_ws, size_t ws_size,
                              hipStream_t stream) {
    const float* z        = (const float*)d_in[0];   // [4,2048,256]
    const float* emb      = (const float*)d_in[1];   // [8192,256]
    const float* code_age = (const float*)d_in[2];   // [8192]
    float* out = (float*)d_out;

    float* enorm   = (float*)d_ws;                            // 8192 f32
    int*   idxbuf  = (int*)((char*)d_ws + (size_t)N_E * 4);   // 8192 i32
    float* partial = (float*)((char*)d_ws + (size_t)2 * N_E * 4); // 8192 f32

    vq_enorm<<<N_E / 8, 256, 0, stream>>>(emb, enorm);

    const size_t ldsBytes = (size_t)(ZL_FLOATS + 2 * EL_FLOATS) * sizeof(float); // ~195 KB
    vq_argmin<<<M_TOK / 128, 256, ldsBytes, stream>>>(z, emb, enorm,
                                                      idxbuf, out + OUT_IDX);

    vq_gather_loss<<<M_TOK, 256, 0, stream>>>(z, emb, idxbuf, out + OUT_ZQ, partial);

    vq_finalize<<<1, 256, 0, stream>>>(partial, code_age, out);
}